// BilinearDecoder_72842645340820
// MI455X (gfx1250) — compile-verified
//
#include <hip/hip_runtime.h>
#include <hip/hip_bf16.h>

#define DD 64
#define NREL 32
#define RT_STRIDE 68   // padded LDS stride for transposed R: even (keeps b64
                       // alignment) and bank-shift 4 => conflict-free B reads

typedef __attribute__((ext_vector_type(2))) float v2f;
typedef __attribute__((ext_vector_type(8))) float v8f;

// ---------------- pre-pass kernels ----------------

__global__ void zero_k(int* counts) {
    if (threadIdx.x < NREL) counts[threadIdx.x] = 0;
}

__global__ void hist_k(const int* __restrict__ rels, int* __restrict__ counts, int n) {
    __shared__ int lcnt[NREL];
    if (threadIdx.x < NREL) lcnt[threadIdx.x] = 0;
    __syncthreads();
    int i = blockIdx.x * blockDim.x + threadIdx.x;
    if (i < n) atomicAdd(&lcnt[rels[i]], 1);
    __syncthreads();
    if (threadIdx.x < NREL && lcnt[threadIdx.x]) atomicAdd(&counts[threadIdx.x], lcnt[threadIdx.x]);
}

__global__ void prefix_k(const int* __restrict__ counts, int* __restrict__ offsets,
                         int* __restrict__ cursor, int* __restrict__ tprefix) {
    if (threadIdx.x == 0) {
        int acc = 0, tacc = 0;
        for (int r = 0; r < NREL; ++r) {
            offsets[r] = acc;
            cursor[r]  = acc;
            tprefix[r] = tacc;
            acc  += counts[r];
            tacc += (counts[r] + 63) >> 6;   // 64-sample tiles per relation
        }
        tprefix[NREL] = tacc;
    }
}

__global__ void scatter_k(const int* __restrict__ rels, int* __restrict__ cursor,
                          int* __restrict__ perm, int n) {
    __shared__ int lcnt[NREL];
    __shared__ int lbase[NREL];
    int t = threadIdx.x;
    if (t < NREL) lcnt[t] = 0;
    __syncthreads();
    int i = blockIdx.x * blockDim.x + t;
    int r = 0, lpos = 0;
    bool ok = (i < n);
    if (ok) { r = rels[i]; lpos = atomicAdd(&lcnt[r], 1); }
    __syncthreads();
    if (t < NREL) lbase[t] = lcnt[t] ? atomicAdd(&cursor[t], lcnt[t]) : 0;
    __syncthreads();
    if (ok) perm[lbase[r] + lpos] = i;
}

// ---------------- main WMMA kernel ----------------
// Block = 128 threads (4 waves). One block handles one 64-sample tile of one
// relation segment. R_rel is staged TRANSPOSED in LDS so each WMMA B-fragment
// is a single contiguous ds_load_b64 into an even VGPR pair.

__global__ void __launch_bounds__(128)
bilinear_wmma_k(const float* __restrict__ e1, const float* __restrict__ e2,
                const float* __restrict__ relE, const int* __restrict__ perm,
                const int* __restrict__ counts, const int* __restrict__ offsets,
                const int* __restrict__ tprefix, float* __restrict__ out) {
    __shared__ __attribute__((aligned(16))) float Rt[DD * RT_STRIDE]; // 17 KB, Rt[e*68+d] = R[d][e]
    __shared__ __attribute__((aligned(16))) float Usm[4][16 * DD];    // 16 KB
    __shared__ int s_map[2];

    const int tid = threadIdx.x;
    if (tid == 0) {
        const int bid = blockIdx.x;
        int rel = -1, tile = 0;
        if (bid < tprefix[NREL]) {
            for (int r = 0; r < NREL; ++r) {
                if (bid < tprefix[r + 1]) { rel = r; tile = bid - tprefix[r]; break; }
            }
        }
        s_map[0] = rel; s_map[1] = tile;
    }
    __syncthreads();
    const int rel = s_map[0];
    if (rel < 0) return;                       // uniform block exit
    const int tile = s_map[1];

    // cooperative load of R_rel (64x64 f32), transposed into LDS
    {
        const float* Rg = relE + (size_t)rel * (DD * DD);
        #pragma unroll
        for (int i = 0; i < 8; ++i) {
            const int f = (tid + i * 128) * 4;     // flat row-major element idx
            const int d = f >> 6;
            const int e = f & 63;
            float4 v = *(const float4*)(Rg + f);   // coalesced 16B global read
            Rt[(e + 0) * RT_STRIDE + d] = v.x;
            Rt[(e + 1) * RT_STRIDE + d] = v.y;
            Rt[(e + 2) * RT_STRIDE + d] = v.z;
            Rt[(e + 3) * RT_STRIDE + d] = v.w;
        }
    }
    __syncthreads();

    const int wave = tid >> 5;
    const int lane = tid & 31;
    const int h    = lane >> 4;    // half-wave select
    const int nn   = lane & 15;

    const int seg  = offsets[rel];
    const int cnt  = counts[rel];
    const int base = tile * 64 + wave * 16;    // sample offset inside segment

    // ---- Stage 1: load A fragments (row m = nn for this lane, clamped tail)
    int pm = base + nn; if (pm > cnt - 1) pm = cnt - 1;
    const int rowA = perm[seg + pm];
    const float* e1row = e1 + (size_t)rowA * DD + 2 * h;

    v2f a[16];
    #pragma unroll
    for (int kb = 0; kb < 16; ++kb) {
        float2 t = *(const float2*)(e1row + 4 * kb);   // A[m][4kb+2h], A[m][4kb+2h+1]
        a[kb].x = t.x; a[kb].y = t.y;
    }

    // ---- accumulate U = A * R over 16 K-chunks x 4 N-tiles
    v8f acc[4];
    #pragma unroll
    for (int n = 0; n < 4; ++n) acc[n] = (v8f){0.f,0.f,0.f,0.f,0.f,0.f,0.f,0.f};

    // B fragment for (kb,n): {R[4kb+2h][16n+nn], R[4kb+2h+1][16n+nn]}
    //                      = contiguous pair Rt[(16n+nn)*68 + 4kb+2h .. +1]
    const float* bbase = &Rt[nn * RT_STRIDE + 2 * h];
    #pragma unroll
    for (int kb = 0; kb < 16; ++kb) {
        #pragma unroll
        for (int n = 0; n < 4; ++n) {
            v2f b = *(const v2f*)(bbase + n * (16 * RT_STRIDE) + 4 * kb);
            acc[n] = __builtin_amdgcn_wmma_f32_16x16x4_f32(
                false, a[kb], false, b, (short)0, acc[n], false, false);
        }
    }

    // ---- spill U (16x64) to LDS: VGPR j holds M = j + 8h, col = 16n+nn
    float* Uw = Usm[wave];
    #pragma unroll
    for (int n = 0; n < 4; ++n)
        #pragma unroll
        for (int j = 0; j < 8; ++j)
            Uw[(j + 8 * h) * DD + 16 * n + nn] = acc[n][j];

    __syncthreads();

    // ---- Stage 2: score_m = U[m,:] . e2[perm_row(m),:]
    if (lane < 16) {
        const int p = base + lane;
        if (p < cnt) {
            const int row = perm[seg + p];
            const float4* e2r = (const float4*)(e2 + (size_t)row * DD);
            const float4* Ur  = (const float4*)(Uw + lane * DD);
            float s = 0.f;
            #pragma unroll
            for (int q = 0; q < 16; ++q) {
                float4 u = Ur[q], v = e2r[q];
                s += u.x * v.x + u.y * v.y + u.z * v.z + u.w * v.w;
            }
            out[row] = s;
        }
    }
}

// ---------------- launcher ----------------

extern "C" void kernel_launch(void* const* d_in, const int* in_sizes, int n_in,
                              void* d_out, int out_size, void* d_ws, size_t ws_size,
                              hipStream_t stream) {
    const float* e1   = (const float*)d_in[0];
    const float* e2   = (const float*)d_in[1];
    const int*   rels = (const int*)d_in[2];
    const float* relE = (const float*)d_in[3];
    float* out = (float*)d_out;
    const int B = in_sizes[2];

    int* ws      = (int*)d_ws;
    int* counts  = ws;            // 32
    int* cursor  = ws + 32;       // 32
    int* offsets = ws + 64;       // 32
    int* tprefix = ws + 96;       // 33
    int* perm    = ws + 256;      // B entries

    const int tb = 256;
    const int nb = (B + tb - 1) / tb;

    zero_k<<<1, 32, 0, stream>>>(counts);
    hist_k<<<nb, tb, 0, stream>>>(rels, counts, B);
    prefix_k<<<1, 1, 0, stream>>>(counts, offsets, cursor, tprefix);
    scatter_k<<<nb, tb, 0, stream>>>(rels, cursor, perm, B);

    const int nblocks = (B + 63) / 64 + NREL;   // upper bound on total tiles
    bilinear_wmma_k<<<nblocks, 128, 0, stream>>>(e1, e2, relE, perm,
                                                 counts, offsets, tprefix, out);
}